// OffsetLoss_42812234006950
// MI455X (gfx1250) — compile-verified
//
#include <hip/hip_runtime.h>
#include <hip/hip_bf16.h>

typedef __attribute__((ext_vector_type(2))) float v2f;
typedef __attribute__((ext_vector_type(4))) float v4f;
typedef __attribute__((ext_vector_type(8))) float v8f;

#define DD 96
#define HH 160
#define WW 160
#define SLICE_ELEMS (DD * HH * WW)       /* 2457600 */
#define SLICE_F4    (SLICE_ELEMS / 4)    /* 614400  */
#define CHUNKS      64
#define F4_PER_CHUNK (SLICE_F4 / CHUNKS) /* 9600    */
#define NSLICES     32                   /* 2 tensors * 16 (b,c) */
#define F4_PER_DHSLAB (HH * WW / 4)      /* 6400 */
#define F4_PER_ROW    (WW / 4)           /* 40   */
#define BETA_F (1.0f / 9.0f)

// Full 32-lane sum. WMMA path: A = {p,0} per lane, B = all-ones.
// Since every B slot is 1.0, D[m][n] = sum_k A[m][k] = p_m + p_{m+16}
// independent of B's lane mapping. C/D layout (documented): lane n in [0,15],
// VGPR r holds D[r][n]; lanes 16-31 hold D[r+8][n-16]. Summing the 8 D VGPRs
// gives half-sums; one xor-16 exchange completes the reduction.
__device__ __forceinline__ float wave_sum32(float p) {
#if __has_builtin(__builtin_amdgcn_wmma_f32_16x16x4_f32)
    v2f a; a[0] = p;    a[1] = 0.0f;
    v2f b; b[0] = 1.0f; b[1] = 1.0f;
    v8f c = {};
    c = __builtin_amdgcn_wmma_f32_16x16x4_f32(
        /*neg_a=*/false, a, /*neg_b=*/false, b,
        /*c_mod=*/(short)0, c, /*reuse_a=*/false, /*reuse_b=*/false);
    float t = ((c[0] + c[1]) + (c[2] + c[3])) + ((c[4] + c[5]) + (c[6] + c[7]));
    t += __shfl_xor(t, 16, 32);
    return t;
#else
    for (int off = 16; off > 0; off >>= 1) p += __shfl_xor(p, off, 32);
    return p;
#endif
}

// Kernel 1: streaming pass. grid = (64 chunks, 32 slices), 256 threads.
// Each block reduces a contiguous 9600-float4 chunk of one (tensor,b,c) slice
// into a 4-float partial {S, Sd, Sh, Sw}.
__global__ __launch_bounds__(256) void k_partials(const float* __restrict__ pred,
                                                  const float* __restrict__ gt,
                                                  float* __restrict__ part) {
    const int chunk = blockIdx.x;       // 0..63
    const int sl    = blockIdx.y;       // 0..31
    const float* base = (sl < 16 ? pred : gt) + (size_t)(sl & 15) * SLICE_ELEMS;
    const v4f* __restrict__ src =
        (const v4f*)base + (size_t)chunk * F4_PER_CHUNK;
    const int qbase = chunk * F4_PER_CHUNK;

    float s = 0.f, sd = 0.f, sh = 0.f, sw = 0.f;
    for (int j = threadIdx.x; j < F4_PER_CHUNK; j += 256) {
        v4f x = __builtin_nontemporal_load(src + j);
        const int q  = qbase + j;               // float4 index within slice
        const int d  = q / F4_PER_DHSLAB;       // 160%4==0, 25600%4==0:
        const int r  = q - d * F4_PER_DHSLAB;   // a float4 never crosses rows
        const int h  = r / F4_PER_ROW;
        const int w0 = (r - h * F4_PER_ROW) * 4;

        const float e0 = __expf(fminf(x.x, 80.0f));
        const float e1 = __expf(fminf(x.y, 80.0f));
        const float e2 = __expf(fminf(x.z, 80.0f));
        const float e3 = __expf(fminf(x.w, 80.0f));
        const float es = (e0 + e1) + (e2 + e3);

        s  += es;
        sd  = fmaf(es, (float)d, sd);
        sh  = fmaf(es, (float)h, sh);
        const float wf = fmaf(3.0f, e3, fmaf(2.0f, e2, e1)); // e1+2e2+3e3
        sw += fmaf(es, (float)w0, wf);
    }

    // cross-lane reduction (full EXEC here: reconverged, uniform flow)
    s  = wave_sum32(s);
    sd = wave_sum32(sd);
    sh = wave_sum32(sh);
    sw = wave_sum32(sw);

    __shared__ float sm[8][4];
    const int wave = threadIdx.x >> 5;
    const int lane = threadIdx.x & 31;
    if (lane == 0) { sm[wave][0] = s; sm[wave][1] = sd; sm[wave][2] = sh; sm[wave][3] = sw; }
    __syncthreads();
    if (threadIdx.x < 4) {
        float acc = 0.f;
        #pragma unroll
        for (int w = 0; w < 8; ++w) acc += sm[w][threadIdx.x];
        part[((size_t)sl * CHUNKS + chunk) * 4 + threadIdx.x] = acc;
    }
}

// Kernel 2: one block per slice; combine 64 partials (double accum) -> point.
__global__ __launch_bounds__(64) void k_points(const float* __restrict__ part,
                                               float* __restrict__ pts) {
    const int sl = blockIdx.x;
    __shared__ float sm[64][4];
    const int t = threadIdx.x;
    const float* p = part + ((size_t)sl * CHUNKS + t) * 4;
    sm[t][0] = p[0]; sm[t][1] = p[1]; sm[t][2] = p[2]; sm[t][3] = p[3];
    __syncthreads();
    if (t == 0) {
        double s = 0.0, sd = 0.0, sh = 0.0, sw = 0.0;
        for (int i = 0; i < CHUNKS; ++i) {
            s += sm[i][0]; sd += sm[i][1]; sh += sm[i][2]; sw += sm[i][3];
        }
        // cx = sum p*(d+1)*D/D - 1 = E[d]; same for h, w.
        pts[sl * 3 + 0] = (float)(sd / s);
        pts[sl * 3 + 1] = (float)(sh / s);
        pts[sl * 3 + 2] = (float)(sw / s);
    }
}

// Kernel 3: smooth-L1 between pred points (slices 0..15) and gt points (16..31).
__global__ void k_loss(const float* __restrict__ pts, float* __restrict__ out) {
    if (threadIdx.x == 0 && blockIdx.x == 0) {
        float acc = 0.f;
        for (int i = 0; i < 16; ++i) {
            #pragma unroll
            for (int k = 0; k < 3; ++k) {
                const float d = fabsf(pts[i * 3 + k] - pts[(16 + i) * 3 + k]);
                acc += (d < BETA_F) ? (0.5f * d * d / BETA_F) : (d - 0.5f * BETA_F);
            }
        }
        out[0] = acc * (1.0f / 16.0f);   // / (B*C)
    }
}

extern "C" void kernel_launch(void* const* d_in, const int* in_sizes, int n_in,
                              void* d_out, int out_size, void* d_ws, size_t ws_size,
                              hipStream_t stream) {
    const float* pred = (const float*)d_in[0];
    const float* gt   = (const float*)d_in[1];
    float* part = (float*)d_ws;                        // 32*64*4 = 8192 floats
    float* pts  = part + (size_t)NSLICES * CHUNKS * 4; // 96 floats

    dim3 g1(CHUNKS, NSLICES);
    k_partials<<<g1, 256, 0, stream>>>(pred, gt, part);
    k_points<<<NSLICES, 64, 0, stream>>>(part, pts);
    k_loss<<<1, 32, 0, stream>>>(pts, (float*)d_out);
}